// RecyclingEmbedder_44890998178025
// MI455X (gfx1250) — compile-verified
//
#include <hip/hip_runtime.h>

typedef __attribute__((ext_vector_type(2))) float v2f;
typedef __attribute__((ext_vector_type(8))) float v8f;

#define N_SEQ 768
#define CM 256
#define CZ 128
#define PPB 16   // pairs per block (16 consecutive j for one i)

// ---------------------------------------------------------------------------
// Kernel 1: m_update = LayerNorm(m).  One wave per row of 256 channels.
// ---------------------------------------------------------------------------
__global__ __launch_bounds__(256) void m_ln_kernel(const float* __restrict__ m,
                                                   const float* __restrict__ g,
                                                   const float* __restrict__ b,
                                                   float* __restrict__ out) {
    const int wave = threadIdx.x >> 5;
    const int lane = threadIdx.x & 31;
    const int row  = blockIdx.x * 8 + wave;
    const float* src = m + (size_t)row * CM;

    float v[8];
    float sum = 0.f;
#pragma unroll
    for (int j = 0; j < 8; ++j) { v[j] = src[lane + 32 * j]; sum += v[j]; }
#pragma unroll
    for (int off = 16; off > 0; off >>= 1) sum += __shfl_xor(sum, off, 32);
    const float mu = sum * (1.f / CM);

    float ss = 0.f;
#pragma unroll
    for (int j = 0; j < 8; ++j) { float d = v[j] - mu; ss += d * d; }
#pragma unroll
    for (int off = 16; off > 0; off >>= 1) ss += __shfl_xor(ss, off, 32);
    const float rs = 1.f / sqrtf(ss * (1.f / CM) + 1e-5f);

    float* dst = out + (size_t)row * CM;
#pragma unroll
    for (int j = 0; j < 8; ++j) {
        const int c = lane + 32 * j;
        dst[c] = (v[j] - mu) * rs * g[c] + b[c];
    }
}

// ---------------------------------------------------------------------------
// Kernel 2: z_update = LN(z) + one_hot(d2 bins) @ W + b,  fused.
//   - 8 KB z tile staged via async global->LDS (ASYNCcnt)
//   - one-hot x W done exactly with V_WMMA_F32_16X16X4_F32 (K padded 15->16)
// ---------------------------------------------------------------------------
__global__ __launch_bounds__(256) void z_kernel(const float* __restrict__ z,
                                                const float* __restrict__ x,
                                                const float* __restrict__ w,
                                                const float* __restrict__ bias,
                                                const float* __restrict__ g,
                                                const float* __restrict__ bt,
                                                float* __restrict__ outz) {
    __shared__ __attribute__((aligned(16))) float sZ[PPB * CZ];  // 8 KB tile
    __shared__ float sMu[PPB];
    __shared__ float sRs[PPB];
    __shared__ int   sIdx[PPB];

    const int tid  = threadIdx.x;
    const int wave = tid >> 5;
    const int lane = tid & 31;
    const long long p0 = (long long)blockIdx.x * PPB;  // first flat pair

    // ---- phase 0: async-copy 16 pairs x 128ch of z into LDS (2 x b128/lane)
    {
        const unsigned base_lds = (unsigned)(size_t)(&sZ[0]);  // flat->LDS offset
#pragma unroll
        for (int it = 0; it < 2; ++it) {
            const int q = tid + it * 256;                       // 16B chunk id
            const unsigned ldsa = base_lds + (unsigned)(q * 16);
            const unsigned voff = (unsigned)(p0 * (CZ * 4) + (long long)q * 16);
            asm volatile("global_load_async_to_lds_b128 %0, %1, %2"
                         :: "v"(ldsa), "v"(voff), "s"(z)
                         : "memory");
        }
    }

    // ---- concurrently: lanes 0-15 of wave 0 compute bin index per pair
    if (tid < PPB) {
        const long long p = p0 + tid;
        const int i = (int)(p / N_SEQ);
        const int j = (int)(p % N_SEQ);
        const float dx = x[i * 3 + 0] - x[j * 3 + 0];
        const float dy = x[i * 3 + 1] - x[j * 3 + 1];
        const float dz = x[i * 3 + 2] - x[j * 3 + 2];
        const float d2 = dx * dx + dy * dy + dz * dz;
        int cnt = 0;
#pragma unroll
        for (int k = 0; k < 15; ++k) {
            const float bk = 3.25f + 1.25f * (float)k;
            cnt += (d2 > bk * bk) ? 1 : 0;
        }
        int idx = 15;  // sentinel: "no bin" -> zero column of padded W
        if (cnt > 0) {
            idx = cnt - 1;
            if (cnt < 15) {  // must also satisfy d2 < upper (strict)
                const float bn = 3.25f + 1.25f * (float)cnt;
                if (d2 >= bn * bn) idx = 15;
            }
        }
        sIdx[tid] = idx;
    }

    asm volatile("s_wait_asynccnt 0x0" ::: "memory");
    __syncthreads();

    // ---- phase 1: LN stats, wave handles pairs 2*wave and 2*wave+1
#pragma unroll
    for (int s = 0; s < 2; ++s) {
        const int mrow = wave * 2 + s;
        const float4 v = ((const float4*)(&sZ[mrow * CZ]))[lane];
        float sum = v.x + v.y + v.z + v.w;
#pragma unroll
        for (int off = 16; off > 0; off >>= 1) sum += __shfl_xor(sum, off, 32);
        const float mu = sum * (1.f / CZ);
        float d0 = v.x - mu, d1 = v.y - mu, d2v = v.z - mu, d3 = v.w - mu;
        float ss = d0 * d0 + d1 * d1 + d2v * d2v + d3 * d3;
#pragma unroll
        for (int off = 16; off > 0; off >>= 1) ss += __shfl_xor(ss, off, 32);
        if (lane == 0) {
            sMu[mrow] = mu;
            sRs[mrow] = 1.f / sqrtf(ss * (1.f / CZ) + 1e-5f);
        }
    }
    __syncthreads();

    // ---- phase 2: 16x16 one-hot x W tile via 4 chained WMMA f32 16x16x4
    // Wave 'wave' owns channels [16*wave, 16*wave+16).
    // ISA layouts: A lane m=lane%16, K = 2*half+{0,1} per VGPR;
    //              B lane n=lane%16, K = 2*half+{0,1};  D: M=r+8*half, N=lane%16.
    const int half = lane >> 4;
    const int nloc = lane & 15;
    const int ch   = wave * 16 + nloc;
    const int myIdx = sIdx[nloc];  // A-row for this lane is pair m = nloc

    v8f acc = {};
#pragma unroll
    for (int c = 0; c < 4; ++c) {
        const int k0 = 4 * c + 2 * half;
        v2f av, bv;
        av.x = (myIdx == k0)     ? 1.f : 0.f;
        av.y = (myIdx == k0 + 1) ? 1.f : 0.f;
        bv.x = (k0     < 15) ? w[ch * 15 + k0]     : 0.f;   // B[k][n] = w[n][k]
        bv.y = (k0 + 1 < 15) ? w[ch * 15 + k0 + 1] : 0.f;   // pad col 15 = 0
        acc = __builtin_amdgcn_wmma_f32_16x16x4_f32(
            /*neg_a=*/false, av, /*neg_b=*/false, bv,
            /*c_mod=*/(short)0, acc, /*reuse_a=*/false, /*reuse_b=*/false);
    }

    const float gch = g[ch], btch = bt[ch], bch = bias[ch];
#pragma unroll
    for (int r = 0; r < 8; ++r) {
        const int mrow = r + 8 * half;
        const float zval = sZ[mrow * CZ + ch];
        const float ln = (zval - sMu[mrow]) * sRs[mrow] * gch + btch;
        __builtin_nontemporal_store(acc[r] + bch + ln,
                                    &outz[(p0 + mrow) * (long long)CZ + ch]);
    }
}

// ---------------------------------------------------------------------------
extern "C" void kernel_launch(void* const* d_in, const int* in_sizes, int n_in,
                              void* d_out, int out_size, void* d_ws, size_t ws_size,
                              hipStream_t stream) {
    const float* m    = (const float*)d_in[0];
    const float* z    = (const float*)d_in[1];
    const float* x    = (const float*)d_in[2];
    const float* w    = (const float*)d_in[3];
    const float* b    = (const float*)d_in[4];
    const float* ln_m_g = (const float*)d_in[5];
    const float* ln_m_b = (const float*)d_in[6];
    const float* ln_z_g = (const float*)d_in[7];
    const float* ln_z_b = (const float*)d_in[8];

    float* out  = (float*)d_out;
    float* outm = out;                        // [768, 256]
    float* outz = out + (size_t)N_SEQ * CM;   // [768, 768, 128]

    m_ln_kernel<<<N_SEQ / 8, 256, 0, stream>>>(m, ln_m_g, ln_m_b, outm);

    const int nblocks = (N_SEQ * N_SEQ) / PPB;  // 36864
    z_kernel<<<nblocks, 256, 0, stream>>>(z, x, w, b, ln_z_g, ln_z_b, outz);
}